// S6_85504208929112
// MI455X (gfx1250) — compile-verified
//
#include <hip/hip_runtime.h>
#include <math.h>

static constexpr int kL      = 16384;
static constexpr int kCH     = 128;
static constexpr int kS      = 32;
static constexpr int kNCHUNK = 64;
static constexpr int kCSTEPS = kL / kNCHUNK;   // 256
static constexpr int kNPAIR  = kCH * kS;       // 4096

typedef __attribute__((ext_vector_type(2))) float v2f;
typedef __attribute__((ext_vector_type(8))) float v8f;

// ---------------------------------------------------------------------------
// Pack weights TRANSPOSED: WpT[n][k], n = 4*s + c with c in {ReWb, ImWb,
// ReWc, ImWc}. Transposed so GEMM B-loads are contiguous along K (b64);
// column order 4s+c so GEMM output rows give one float4 (Br,Bi,Cr,Ci) per s.
// ---------------------------------------------------------------------------
__global__ void s6_pack_w(const float2* __restrict__ Wb, const float2* __restrict__ Wc,
                          float* __restrict__ WpT) {
    int i = blockIdx.x * blockDim.x + threadIdx.x;   // 0 .. 128*128-1
    int n = i >> 7;          // output column 0..127
    int k = i & 127;         // input channel
    int s = n >> 2;
    int c = n & 3;
    float2 wv = (c < 2) ? Wb[k * kS + s] : Wc[k * kS + s];
    float v = (c & 1) ? wv.y : wv.x;
    WpT[n * kCH + k] = v;
}

// ---------------------------------------------------------------------------
// u[t] = x[t,:] . W1 + b1   (tiny GEMV)
// ---------------------------------------------------------------------------
__global__ void s6_gemv_u(const float* __restrict__ X, const float* __restrict__ W1,
                          const float* __restrict__ b1, float* __restrict__ u) {
    int t = blockIdx.x * blockDim.x + threadIdx.x;
    if (t >= kL) return;
    float acc = 0.0f;
    #pragma unroll
    for (int c = 0; c < kCH; ++c) acc += X[t * kCH + c] * W1[c];
    u[t] = acc + b1[0];
}

// ---------------------------------------------------------------------------
// G = X @ WpT^T  with V_WMMA_F32_16X16X4_F32.
// One wave per 16x16 output tile; block = 8 waves covering all 8 N-tiles,
// blockIdx.x = M tile (L/16 = 1024 blocks).
// A layout (16x4): lane m=lane&15, elem j -> A[m][2*half + j]
// B layout (4x16): lane n=lane&15, elem j -> B[2*half + j][n] = WpT[n][2*half+j]
// D layout (16x16): elem j -> D[j + 8*half][n]
// ---------------------------------------------------------------------------
__global__ void s6_gemm(const float* __restrict__ X, const float* __restrict__ WT,
                        float* __restrict__ OUT) {
    const int wave = threadIdx.x >> 5;   // 0..7 -> N tile
    const int lane = threadIdx.x & 31;
    const int half = lane >> 4;
    const int l16  = lane & 15;
    const int m0   = blockIdx.x * 16;
    const int n0   = wave * 16;

    const float* xrow = X  + (m0 + l16) * kCH + 2 * half;   // along K
    const float* wrow = WT + (n0 + l16) * kCH + 2 * half;   // along K (transposed W)

    v8f acc = {};
    #pragma unroll
    for (int k = 0; k < kCH; k += 4) {
        v2f a; a.x = xrow[k]; a.y = xrow[k + 1];
        v2f b; b.x = wrow[k]; b.y = wrow[k + 1];
        acc = __builtin_amdgcn_wmma_f32_16x16x4_f32(
            /*neg_a=*/false, a, /*neg_b=*/false, b,
            /*c_mod=*/(short)0, acc, /*reuse_a=*/false, /*reuse_b=*/false);
    }
    #pragma unroll
    for (int j = 0; j < 8; ++j)
        OUT[(m0 + j + 8 * half) * kCH + n0 + l16] = acc[j];
}

// ---------------------------------------------------------------------------
__device__ __forceinline__ float s6_softplus(float z) {
    return (z > 0.0f) ? (z + log1pf(expf(-z))) : log1pf(expf(z));
}

// ---------------------------------------------------------------------------
// Scan kernel. One wave per (chunk, channel); lane = state index s.
// FINAL=false: compute chunk summaries P = prod(At), U = h_local (h0 = 0).
// FINAL=true : start from Hinit[chunk], emit y[t,ch] = 2*Re(sum_s C*h)
//              via wave32 butterfly reduction.
// G rows are laid out as [t][s][4] = (Br,Bi,Cr,Ci): one b128 load per step.
// ---------------------------------------------------------------------------
template <bool FINAL>
__global__ void s6_scan(const float* __restrict__ X, const float* __restrict__ G,
                        const float* __restrict__ uArr,
                        const float* __restrict__ lognegA, const float* __restrict__ Aimag,
                        const float2* __restrict__ bb, const float2* __restrict__ bc,
                        const float* __restrict__ W2, const float* __restrict__ b2,
                        float2* __restrict__ Pbuf, float2* __restrict__ Ubuf,
                        const float2* __restrict__ Hinit,
                        float* __restrict__ out) {
    const int s  = threadIdx.x & 31;                 // lane = state index
    const int gw = blockIdx.x * (blockDim.x >> 5) + (threadIdx.x >> 5);
    // force wave-uniformity so x[t,ch] / u[t] can use the scalar (SMEM) path
    const int gwu  = __builtin_amdgcn_readfirstlane(gw);
    const int c    = gwu / kCH;                      // chunk
    const int ch   = gwu % kCH;                      // channel
    const int pidx = ch * kS + s;

    const float dtb = logf(expf(0.01f) - 1.0f);

    const float Ar  = -expf(lognegA[pidx]);
    const float Ai  = Aimag[pidx];
    const float den = Ar * Ar + Ai * Ai;
    const float iAr =  Ar / den;
    const float iAi = -Ai / den;
    const float2 bbv = bb[s];
    const float2 bcv = bc[s];
    const float  w2c = W2[ch];
    const float  b2c = b2[ch];

    float hr, hi;
    if (FINAL) {
        float2 h0 = Hinit[c * kNPAIR + pidx];
        hr = h0.x; hi = h0.y;
    } else {
        hr = 0.0f; hi = 0.0f;
    }
    float Pr = 1.0f, Pi = 0.0f;

    const int t0 = c * kCSTEPS;
    for (int tt = 0; tt < kCSTEPS; ++tt) {
        const int t  = t0 + tt;
        const float xv = X[t * kCH + ch];            // wave-uniform
        const float uv = uArr[t];                    // wave-uniform
        const float dt = s6_softplus(dtb + uv * w2c + b2c);

        // At = exp(A * dt)
        const float er = expf(Ar * dt);
        float sn, cs;
        __sincosf(Ai * dt, &sn, &cs);
        const float atr = er * cs;
        const float ati = er * sn;

        // (Br,Bi,Cr,Ci) in one 16B load
        const float4 g4 = *(const float4*)(G + (size_t)t * kCH + s * 4);
        const float Br = 1.0f + g4.x + bbv.x;
        const float Bi =        g4.y + bbv.y;
        const float Cr =        g4.z + bcv.x;
        const float Ci =        g4.w + bcv.y;

        // ut = (At - 1) * (1/A) * B * x
        const float dr  = atr - 1.0f, di = ati;
        const float m1r = dr * iAr - di * iAi;
        const float m1i = dr * iAi + di * iAr;
        const float m2r = m1r * Br - m1i * Bi;
        const float m2i = m1r * Bi + m1i * Br;
        const float ur  = m2r * xv;
        const float ui  = m2i * xv;

        // h = At * h + ut
        const float nhr = atr * hr - ati * hi + ur;
        const float nhi = atr * hi + ati * hr + ui;
        hr = nhr; hi = nhi;

        if (FINAL) {
            float yv = Cr * hr - Ci * hi;
            #pragma unroll
            for (int off = 16; off > 0; off >>= 1)
                yv += __shfl_xor(yv, off, 32);
            if (s == 0) out[t * kCH + ch] = 2.0f * yv;
        } else {
            const float nPr = atr * Pr - ati * Pi;
            const float nPi = atr * Pi + ati * Pr;
            Pr = nPr; Pi = nPi;
        }
    }

    if (!FINAL) {
        float2 pv;  pv.x  = Pr; pv.y  = Pi;
        float2 uvv; uvv.x = hr; uvv.y = hi;
        Pbuf[c * kNPAIR + pidx] = pv;
        Ubuf[c * kNPAIR + pidx] = uvv;
    }
}

// ---------------------------------------------------------------------------
// Sequential carry across chunks: Hinit[c] = h before chunk c.
// One thread per (ch, s) pair.
// ---------------------------------------------------------------------------
__global__ void s6_fix(const float2* __restrict__ Pbuf, const float2* __restrict__ Ubuf,
                       float2* __restrict__ Hinit) {
    const int idx = blockIdx.x * blockDim.x + threadIdx.x;  // 0..4095
    float hr = 0.0f, hi = 0.0f;
    for (int c = 0; c < kNCHUNK; ++c) {
        float2 h; h.x = hr; h.y = hi;
        Hinit[c * kNPAIR + idx] = h;
        const float2 P = Pbuf[c * kNPAIR + idx];
        const float2 U = Ubuf[c * kNPAIR + idx];
        const float nhr = P.x * hr - P.y * hi + U.x;
        const float nhi = P.x * hi + P.y * hr + U.y;
        hr = nhr; hi = nhi;
    }
}

// ---------------------------------------------------------------------------
extern "C" void kernel_launch(void* const* d_in, const int* in_sizes, int n_in,
                              void* d_out, int out_size, void* d_ws, size_t ws_size,
                              hipStream_t stream) {
    (void)in_sizes; (void)n_in; (void)out_size; (void)ws_size;

    const float*  x       = (const float*)d_in[0];
    const float*  lognegA = (const float*)d_in[1];
    const float*  Aimag   = (const float*)d_in[2];
    const float2* Wb      = (const float2*)d_in[3];
    const float2* bb      = (const float2*)d_in[4];
    const float2* Wc      = (const float2*)d_in[5];
    const float2* bc      = (const float2*)d_in[6];
    const float*  W1      = (const float*)d_in[7];
    const float*  b1      = (const float*)d_in[8];
    const float*  W2      = (const float*)d_in[9];
    const float*  b2      = (const float*)d_in[10];
    float* out = (float*)d_out;

    // Workspace layout (floats); G is 16B-aligned (131072-byte offset).
    float*  Wp = (float*)d_ws;                       // 128*128
    float*  u  = Wp + kCH * kCH;                     // L
    float*  G  = u + kL;                             // L*128
    float2* P  = (float2*)(G + (size_t)kL * kCH);    // NCHUNK*4096 complex
    float2* U  = P + (size_t)kNCHUNK * kNPAIR;
    float2* H  = U + (size_t)kNCHUNK * kNPAIR;

    // 1. pack weights (transposed + interleaved columns)
    s6_pack_w<<<(kCH * kCH) / 256, 256, 0, stream>>>(Wb, Wc, Wp);
    // 2. u = x @ W1 + b1
    s6_gemv_u<<<kL / 256, 256, 0, stream>>>(x, W1, b1, u);
    // 3. G = x @ W  (WMMA f32 16x16x4)
    s6_gemm<<<kL / 16, 256, 0, stream>>>(x, Wp, G);
    // 4. chunk summaries
    s6_scan<false><<<(kNCHUNK * kCH) / 8, 256, 0, stream>>>(
        x, G, u, lognegA, Aimag, bb, bc, W2, b2, P, U, H, out);
    // 5. carry across chunks
    s6_fix<<<kNPAIR / 256, 256, 0, stream>>>(P, U, H);
    // 6. final scan + output
    s6_scan<true><<<(kNCHUNK * kCH) / 8, 256, 0, stream>>>(
        x, G, u, lognegA, Aimag, bb, bc, W2, b2, P, U, H, out);
}